// CustomNMS_26061861552412
// MI455X (gfx1250) — compile-verified
//
#include <hip/hip_runtime.h>

#define PRE 256
#define POST 128
#define TH_SCORE 0.1f
#define TH_NMS 0.2f

typedef float v2f __attribute__((ext_vector_type(2)));
typedef float v8f __attribute__((ext_vector_type(8)));

// monotone map float -> u32 so that unsigned compare == float compare
__device__ __forceinline__ unsigned sortKeyHi(float s) {
  unsigned b = __float_as_uint(s);
  return (b & 0x80000000u) ? ~b : (b | 0x80000000u);
}
__device__ __forceinline__ float decodeScore(unsigned u) {
  unsigned b = (u & 0x80000000u) ? (u & 0x7FFFFFFFu) : ~u;
  return __uint_as_float(b);
}

__global__ __launch_bounds__(256) void rnms_kernel(const float* __restrict__ boxes,
                                                   const int* __restrict__ nbox_arr,
                                                   float* __restrict__ out, int N) {
  const int b   = blockIdx.x;
  const int tid = threadIdx.x;
  const float* fb = boxes + (size_t)b * N * 9;
  float*       fo = out   + (size_t)b * POST * 9;
  const int nbox  = nbox_arr[b];

  __shared__ unsigned long long redK[256]; // argmax reduction keys
  __shared__ int      selIdx[PRE];
  __shared__ float    selScore[PRE];
  __shared__ float    lab1[PRE];
  __shared__ float    crn[PRE][8];         // 4 corners (x,y) CCW
  __shared__ float    ctrx[PRE], ctry[PRE];
  __shared__ float    nrm[PRE];            // |c|^2
  __shared__ float    radS[PRE];           // half-diagonal
  __shared__ float    areaS[PRE];
  __shared__ unsigned candM[PRE][8];       // coarse circle-overlap candidates
  __shared__ unsigned supM[PRE][8];        // iou>thresh, j>i only
  __shared__ unsigned keepM[8];

  // ---- phase 0: zero output + masks, prefetch frame into cache hierarchy
  for (int k = tid; k < POST * 9; k += 256) fo[k] = 0.0f;
  for (int k = tid; k < PRE * 8; k += 256) { candM[k >> 3][k & 7] = 0u; supM[k >> 3][k & 7] = 0u; }
  {
    const char* base = (const char*)fb;
    size_t bytes = (size_t)N * 9 * 4;
    for (size_t off = (size_t)tid * 128; off < bytes; off += 256 * 128)
      __builtin_prefetch(base + off, 0, 1);
  }
  __syncthreads();

  // ---- phase 1: top-256 selection, strictly-decreasing key argmax (no scratch)
  unsigned long long prev = ~0ULL;
  for (int r = 0; r < PRE; ++r) {
    unsigned long long best = 0ULL;
    for (int k = tid; k < N; k += 256) {
      float s = fb[(size_t)k * 9 + 8];
      bool valid = (k < nbox) && (s > TH_SCORE);
      unsigned u = sortKeyHi(valid ? s : -__builtin_inff());
      unsigned long long key = ((unsigned long long)u << 32) | (unsigned)(~k);
      if (key < prev && key > best) best = key;
    }
    redK[tid] = best;
    __syncthreads();
    for (int st = 128; st > 0; st >>= 1) {
      if (tid < st) { unsigned long long o = redK[tid + st]; if (o > redK[tid]) redK[tid] = o; }
      __syncthreads();
    }
    unsigned long long win = redK[0];
    if (tid == 0) {
      selIdx[r]   = (int)(~(unsigned)(win & 0xFFFFFFFFu));
      selScore[r] = decodeScore((unsigned)(win >> 32));
    }
    prev = win;
    __syncthreads();
  }

  // ---- phase 2: per-box geometry (one thread per selected box)
  {
    int i = tid;
    int idx = selIdx[i];
    const float* row = fb + (size_t)idx * 9;
    float x = row[0], y = row[1], dx = row[3], dy = row[4], r = row[6];
    lab1[i] = row[7] + 1.0f;
    float c = cosf(r), s = sinf(r);
    const float tx[4] = {0.5f, -0.5f, -0.5f, 0.5f};
    const float ty[4] = {0.5f, 0.5f, -0.5f, -0.5f};
    #pragma unroll
    for (int k = 0; k < 4; ++k) {
      float px = tx[k] * dx, py = ty[k] * dy;
      crn[i][2 * k]     = c * px - s * py + x;
      crn[i][2 * k + 1] = s * px + c * py + y;
    }
    ctrx[i] = x; ctry[i] = y; nrm[i] = x * x + y * y;
    areaS[i] = dx * dy;
    radS[i] = 0.5f * sqrtf(dx * dx + dy * dy);
  }
  __syncthreads();

  // ---- phase 3: WMMA Gram matrix C*C^T (rank-2, K padded to 4) -> coarse filter
  {
    const int wave = tid >> 5;
    const int lane = tid & 31;
    const int l15  = lane & 15;
    const float msk = (lane < 16) ? 1.0f : 0.0f; // lanes 16-31 carry K=2,3 (zero)
    for (int t = wave; t < 256; t += 8) {        // 16x16 tile grid, uniform per wave
      int ti = t >> 4, tj = t & 15;
      int m0 = ti * 16 + l15, n0 = tj * 16 + l15;
      v2f a  = { ctrx[m0] * msk, ctry[m0] * msk };  // A: 16x4 (K=0,1 in v0,v1)
      v2f bv = { ctrx[n0] * msk, ctry[n0] * msk };  // B: 4x16 row-striped
      v8f cacc = {};
      v8f d = __builtin_amdgcn_wmma_f32_16x16x4_f32(
          /*neg_a=*/false, a, /*neg_b=*/false, bv,
          /*c_mod=*/(short)0, cacc, /*reuse_a=*/false, /*reuse_b=*/false);
      int wrd = tj >> 1, sh = (tj & 1) * 16;
      #pragma unroll
      for (int v = 0; v < 8; ++v) {
        int m = ti * 16 + v + ((lane >= 16) ? 8 : 0);
        int n = tj * 16 + l15;
        float dist2 = nrm[m] + nrm[n] - 2.0f * d[v];
        float rs = radS[m] + radS[n];
        bool cd = dist2 <= rs * rs + 1e-3f;
        unsigned bal = __builtin_amdgcn_ballot_w32(cd);
        if (lane == 0) {
          atomicOr(&candM[ti * 16 + v][wrd],     (bal & 0xFFFFu) << sh);
          atomicOr(&candM[ti * 16 + v + 8][wrd], (bal >> 16)     << sh);
        }
      }
    }
  }
  __syncthreads();

  // ---- phase 4: exact rotated intersection (Sutherland-Hodgman) on candidates
  for (int p = tid; p < PRE * PRE; p += 256) {
    int i = p >> 8, j = p & 255;
    if (j <= i) continue;
    if (!((candM[i][j >> 5] >> (j & 31)) & 1u)) continue;
    float px[8], py[8]; int n = 4;
    #pragma unroll
    for (int k = 0; k < 4; ++k) { px[k] = crn[i][2 * k]; py[k] = crn[i][2 * k + 1]; }
    for (int e = 0; e < 4 && n > 0; ++e) {
      float ax = crn[j][2 * e], ay = crn[j][2 * e + 1];
      int e1 = (e + 1) & 3;
      float ex = crn[j][2 * e1] - ax, ey = crn[j][2 * e1 + 1] - ay;
      float qx[8], qy[8]; int m2 = 0;
      for (int k = 0; k < n; ++k) {
        int k1 = (k + 1 == n) ? 0 : k + 1;
        float dc = ex * (py[k]  - ay) - ey * (px[k]  - ax);
        float dn = ex * (py[k1] - ay) - ey * (px[k1] - ax);
        if (dc >= 0.0f) { qx[m2] = px[k]; qy[m2] = py[k]; ++m2; }
        if ((dc > 0.0f && dn < 0.0f) || (dc < 0.0f && dn > 0.0f)) {
          float tpar = dc / (dc - dn);
          qx[m2] = px[k] + tpar * (px[k1] - px[k]);
          qy[m2] = py[k] + tpar * (py[k1] - py[k]);
          ++m2;
        }
      }
      n = m2;
      for (int k = 0; k < n; ++k) { px[k] = qx[k]; py[k] = qy[k]; }
    }
    float inter = 0.0f;
    if (n >= 3) {
      float acc = 0.0f;
      for (int k = 0; k < n; ++k) {
        int k1 = (k + 1 == n) ? 0 : k + 1;
        acc += px[k] * py[k1] - py[k] * px[k1];
      }
      inter = 0.5f * fabsf(acc);
    }
    float uni = areaS[i] + areaS[j] - inter;
    float iou = inter / fmaxf(uni, 1e-6f);
    if (iou > TH_NMS) atomicOr(&supM[i][j >> 5], 1u << (j & 31));
  }
  __syncthreads();

  // ---- phase 5: sequential greedy suppression (bitmask, one lane)
  if (tid == 0) {
    unsigned km[8];
    #pragma unroll
    for (int w = 0; w < 8; ++w) km[w] = 0u;
    for (int i = 0; i < PRE; ++i)
      if (selScore[i] > -__builtin_inff()) km[i >> 5] |= 1u << (i & 31);
    for (int i = 0; i < PRE; ++i) {
      if ((km[i >> 5] >> (i & 31)) & 1u) {
        #pragma unroll
        for (int w = 0; w < 8; ++w) km[w] &= ~supM[i][w];
      }
    }
    #pragma unroll
    for (int w = 0; w < 8; ++w) keepM[w] = km[w];
  }
  __syncthreads();

  // ---- phase 6: compact survivors into first 128 output rows
  {
    int i = tid;
    unsigned w = i >> 5, bit = i & 31;
    if ((keepM[w] >> bit) & 1u) {
      int rank = 0;
      for (unsigned ww = 0; ww < w; ++ww) rank += __popc(keepM[ww]);
      rank += __popc(keepM[w] & ((1u << bit) - 1u));
      if (rank < POST) {
        const float* row = fb + (size_t)selIdx[i] * 9;
        float* o = fo + rank * 9;
        #pragma unroll
        for (int c = 0; c < 7; ++c) o[c] = row[c];
        o[7] = selScore[i];
        o[8] = lab1[i];
      }
    }
  }
}

extern "C" void kernel_launch(void* const* d_in, const int* in_sizes, int n_in,
                              void* d_out, int out_size, void* d_ws, size_t ws_size,
                              hipStream_t stream) {
  const float* boxes = (const float*)d_in[0]; // [B, N, 9]
  const int*   nbox  = (const int*)d_in[1];   // [B]
  float*       out   = (float*)d_out;         // [B, POST, 9]
  int B = in_sizes[1];
  int N = (B > 0) ? in_sizes[0] / (B * 9) : 0;
  if (B <= 0 || N <= 0) return;
  rnms_kernel<<<B, 256, 0, stream>>>(boxes, nbox, out, N);
}